// SpatialGradient_8950711845596
// MI455X (gfx1250) — compile-verified
//
#include <hip/hip_runtime.h>
#include <stdint.h>

// ---------------------------------------------------------------------------
// Sobel spatial gradient, [8,32,512,512] fp32, depthwise 3x3, SAME zero pad.
// HBM-bound (AI ~5 flop/byte vs 23.3 TB/s) -> optimize data movement:
//   global -> LDS via CDNA5 async-to-LDS DMA (ASYNCcnt, no VGPR staging),
//   compute from LDS, 128-bit non-temporal stores for the streaming output.
// 128x32 tiles: staged halo overhead ~13%, 16384 blocks, 18.5 KB LDS/block.
// ---------------------------------------------------------------------------

typedef float v4f __attribute__((ext_vector_type(4)));

#define AS_LOCAL __attribute__((address_space(3)))

__device__ __forceinline__ void async_load_b128(const float* g, float* l) {
    // LDS byte offset (AS3 pointers are 32-bit offsets into LDS).
    unsigned loff = (unsigned)(uintptr_t)(AS_LOCAL float*)l;
    unsigned long long ga = (unsigned long long)(uintptr_t)g;
    asm volatile("global_load_async_to_lds_b128 %0, %1, off"
                 :: "v"(loff), "v"(ga) : "memory");
}

__device__ __forceinline__ void wait_async_zero() {
#if __has_builtin(__builtin_amdgcn_s_wait_asynccnt)
    __builtin_amdgcn_s_wait_asynccnt(0);
#else
    asm volatile("s_wait_asynccnt 0" ::: "memory");
#endif
}

namespace {
constexpr int Himg = 512;
constexpr int Wimg = 512;
constexpr int TW   = 128;   // output tile width
constexpr int TH   = 32;    // output tile height
constexpr int LW   = 136;   // LDS row: TW + 8 (one 16B segment of slack/side)
constexpr int NROW = TH + 2;                 // 34 rows incl. halo
constexpr int NSEG = LW / 4;                 // 34 b128 segments per row
constexpr int NLOAD = NROW * NSEG;           // 1156 segments per tile
}

__global__ __launch_bounds__(256)
void sobel_grad_kernel(const float* __restrict__ in,
                       const int*   __restrict__ dirp,
                       float*       __restrict__ out) {
    __shared__ __align__(16) float tile[NROW][LW];   // 18496 B of 320 KB/WGP

    const int tid = threadIdx.x;
    const int tx0 = blockIdx.x * TW;
    const int ty0 = blockIdx.y * TH;
    const size_t planeOff = (size_t)blockIdx.z * Himg * Wimg;

    // ---- Stage tile (with halo) into LDS via async DMA; zero-fill OOB halo.
    // LDS row covers global cols [tx0-4, tx0+132); tiles are 128-aligned so
    // every 16B segment is fully in-range or fully out-of-range.
    for (int idx = tid; idx < NLOAD; idx += 256) {
        const int r  = idx / NSEG;
        const int s  = idx - r * NSEG;
        const int gy = ty0 - 1 + r;
        const int gx = tx0 - 4 + s * 4;
        float* ldst = &tile[r][s * 4];
        if ((unsigned)gy < (unsigned)Himg && (unsigned)gx < (unsigned)Wimg) {
            async_load_b128(in + planeOff + (size_t)gy * Wimg + gx, ldst);
        } else {
            v4f z = {0.f, 0.f, 0.f, 0.f};           // SAME zero padding
            *(v4f*)ldst = z;
        }
    }
    wait_async_zero();      // own wave's async copies landed in LDS
    __syncthreads();        // all waves' copies visible

    const int dir = *dirp;  // uniform scalar load

    // ---- Each thread: four 4-wide output strips (rows r0, r0+8, +16, +24).
    const int r0 = tid >> 5;          // 0..7
    const int j0 = (tid & 31) << 2;   // 0,4,...,124
    const int cb = j0 + 3;            // LDS col of (x-1) for first output

#pragma unroll
    for (int rr = 0; rr < 4; ++rr) {
        const int r = r0 + rr * 8;    // output row within tile, 0..31
        const float* ra = &tile[r    ][cb];   // y-1
        const float* rb = &tile[r + 1][cb];   // y
        const float* rc = &tile[r + 2][cb];   // y+1
        float a[6], b[6], c[6];
#pragma unroll
        for (int k = 0; k < 6; ++k) { a[k] = ra[k]; b[k] = rb[k]; c[k] = rc[k]; }

        v4f res;
#pragma unroll
        for (int k = 0; k < 4; ++k) {
            // cross-correlation with kx=[[-1,0,1],[-2,0,2],[-1,0,1]],
            //                        ky=[[-1,-2,-1],[0,0,0],[1,2,1]]
            const float gx = (a[k+2] - a[k]) + 2.0f * (b[k+2] - b[k]) + (c[k+2] - c[k]);
            const float gy = (c[k]   - a[k]) + 2.0f * (c[k+1] - a[k+1]) + (c[k+2] - a[k+2]);
            float v;
            if      (dir < 0) v = __builtin_fabsf(gx);
            else if (dir > 0) v = __builtin_fabsf(gy);
            else              v = __builtin_sqrtf(gx * gx + gy * gy);
            res[k] = v;
        }
        // Streaming, write-once output: non-temporal 128-bit store.
        float* dst = out + planeOff + (size_t)(ty0 + r) * Wimg + tx0 + j0;
        __builtin_nontemporal_store(res, (v4f*)dst);
    }
}

extern "C" void kernel_launch(void* const* d_in, const int* in_sizes, int n_in,
                              void* d_out, int out_size, void* d_ws, size_t ws_size,
                              hipStream_t stream) {
    (void)in_sizes; (void)n_in; (void)out_size; (void)d_ws; (void)ws_size;
    const float* in  = (const float*)d_in[0];
    const int*   dir = (const int*)d_in[1];
    float*       out = (float*)d_out;

    dim3 grid(Wimg / TW, Himg / TH, 8 * 32);   // (4, 16, 256)
    sobel_grad_kernel<<<grid, 256, 0, stream>>>(in, dir, out);
}